// DiscreteKeyValueBottleneck_86663850098809
// MI455X (gfx1250) — compile-verified
//
#include <hip/hip_runtime.h>

// DiscreteKeyValueBottleneck gather:
//   out[b,n,h*12+d] = values[h, idx[b,n,h], d]
// B=64, N=2048, H=64, C=4096, D=12. Pure gather, 0 FLOPs.
// Roofline: ~403 MB NT output stream + ~34 MB NT index stream + 12.6 MB
// table (L2-resident, 192 MB L2) => ~19 us at 23.3 TB/s. Memory-bound;
// WMMA not applicable. Optimization = coalescing + cache-policy control.

typedef float v4f __attribute__((ext_vector_type(4)));

constexpr unsigned kB = 64, kN = 2048, kH = 64, kC = 4096, kD = 12;
// One float4 of output per thread: 3 float4s per gathered row.
constexpr unsigned kTotalV4 = kB * kN * kH * (kD / 4);   // 25,165,824
constexpr unsigned kBlock   = 256;                       // 8 waves / WGP pass

__global__ __launch_bounds__(kBlock)
void dkvb_gather_kernel(const int* __restrict__ indices,    // (B,N,H) int32
                        const float* __restrict__ values,   // (H,C,D) f32
                        float* __restrict__ out) {          // (B,N,H*D) f32
    const unsigned g = blockIdx.x * kBlock + threadIdx.x;   // < 25.2M, fits u32
    const unsigned e = g / 3u;            // element id: ((b*N + n)*H + h)
    const unsigned j = g - e * 3u;        // which float4 of the 12-float row
    const unsigned h = e & (kH - 1u);     // H == 64, power of two

    // Per-wave prefetch of the index stream ~43 KB ahead (hides HBM latency
    // on the only cold streaming read). Lowers to global_prefetch_b8.
    if ((threadIdx.x & 31u) == 0u) {
        __builtin_prefetch(indices + e + 10880u, 0, 0);
    }

    // Index stream is read exactly once -> non-temporal load.
    const int idx = __builtin_nontemporal_load(indices + e);

    // Row start = (h*C + idx)*12 floats = 48 B aligned -> every float4 chunk
    // is 16 B aligned. Table is 12.6 MB: keep regular-temporal so it stays
    // L2-resident while the output stream bypasses with NT.
    const v4f* __restrict__ src =
        (const v4f*)(values + ((size_t)h * kC + (unsigned)idx) * kD) + j;
    const v4f v = *src;

    // Output written once, never re-read -> non-temporal b128 store.
    // Consecutive lanes -> consecutive 16 B chunks: fully coalesced.
    __builtin_nontemporal_store(v, (v4f*)out + g);
}

extern "C" void kernel_launch(void* const* d_in, const int* in_sizes, int n_in,
                              void* d_out, int out_size, void* d_ws, size_t ws_size,
                              hipStream_t stream) {
    (void)in_sizes; (void)n_in; (void)d_ws; (void)ws_size; (void)out_size;

    const int*   indices = (const int*)d_in[0];    // memory_indices (B,N,H)
    const float* values  = (const float*)d_in[1];  // values (H,C,D)
    float*       out     = (float*)d_out;          // (B, N, H*D)

    static_assert(kTotalV4 % kBlock == 0, "exact grid, no tail");
    dim3 grid(kTotalV4 / kBlock);
    dim3 block(kBlock);
    dkvb_gather_kernel<<<grid, block, 0, stream>>>(indices, values, out);
}